// involution_335007449667
// MI455X (gfx1250) — compile-verified
//
#include <hip/hip_runtime.h>

typedef __attribute__((ext_vector_type(2))) float v2f;
typedef __attribute__((ext_vector_type(8))) float v8f;

#define CIN 256
#define RED 64
#define HW  3136
#define W56 56
#define KK  49      // 7x7 kernel footprint
#define NG  16      // groups
#define GC  16      // channels per group
#define BN_EPS 1e-5f

// ---------------------------------------------------------------------------
// Kernel 1: t[b,64,hw] = relu(bn1(w1[64,256] @ x[b,256,hw]))
// WMMA f32 16x16x4. Block = 8 waves: 4 M-tiles (RED=64) x 2 N-tiles (32 px).
// ---------------------------------------------------------------------------
__global__ __launch_bounds__(256) void conv1_bn_relu_kernel(
    const float* __restrict__ x, const float* __restrict__ w1,
    const float* __restrict__ g1, const float* __restrict__ be1,
    const float* __restrict__ mu1, const float* __restrict__ va1,
    float* __restrict__ t)
{
  const int b    = blockIdx.y;
  const int lane = threadIdx.x & 31;
  const int wave = threadIdx.x >> 5;
  const int mt   = wave & 3;             // which 16 rows of RED
  const int nt   = wave >> 2;            // which 16 pixels of the 32-px tile
  const int p_base = blockIdx.x * 32 + nt * 16;
  const int l15    = lane & 15;
  const int khalf  = (lane >> 4) << 1;   // lanes 16-31 hold K=2,3 (ISA 7.12.2)
  const int row    = mt * 16 + l15;      // A: M = lane&15

  const float* wrow = w1 + row * CIN;
  const float* xb   = x + (size_t)b * CIN * HW + p_base + l15;

  v8f acc = {};
  #pragma unroll 4
  for (int k = 0; k < CIN; k += 4) {
    const float2 av = *(const float2*)(wrow + k + khalf);   // contiguous K pair
    v2f A; A.x = av.x; A.y = av.y;
    v2f B;
    B.x = xb[(size_t)(k + khalf)     * HW];
    B.y = xb[(size_t)(k + khalf + 1) * HW];
    acc = __builtin_amdgcn_wmma_f32_16x16x4_f32(
        false, A, false, B, (short)0, acc, false, false);
  }

  // D layout: VGPR r, lane L -> M = r + 8*(L>=16), N = L&15
  float* tb = t + (size_t)b * RED * HW + p_base + l15;
  #pragma unroll
  for (int r = 0; r < 8; ++r) {
    const int o   = mt * 16 + r + ((lane >> 4) << 3);
    const float inv = g1[o] * rsqrtf(va1[o] + BN_EPS);
    const float sh  = be1[o] - mu1[o] * inv;
    tb[(size_t)o * HW] = fmaxf(acc[r] * inv + sh, 0.f);
  }
}

// ---------------------------------------------------------------------------
// Kernel 2 (fused): per 16-pixel tile
//   wgt[784,16] = w2[784,64] @ t_tile[64,16] + b2      (WMMA, kept in LDS)
//   out[c,p]    = relu(bn2( sum_k wgt[g*49+k,p] * x[g*16+c, nbhd_k(p)] ))
// Fusing avoids materializing the 78.7 MB wgt tensor in HBM.
// ---------------------------------------------------------------------------
__global__ __launch_bounds__(256) void conv2_involution_kernel(
    const float* __restrict__ x,  const float* __restrict__ t,
    const float* __restrict__ w2, const float* __restrict__ b2,
    const float* __restrict__ g2, const float* __restrict__ be2,
    const float* __restrict__ mu2, const float* __restrict__ va2,
    float* __restrict__ out)
{
  __shared__ float t_lds[RED * 16];          //  4 KB
  __shared__ float wgt_lds[KK * NG * 16];    // 50 KB (784 x 16)

  const int b    = blockIdx.y;
  const int p0   = blockIdx.x * 16;
  const int tid  = threadIdx.x;
  const int lane = tid & 31;
  const int wave = tid >> 5;
  const int l15  = lane & 15;

  // Stage the 64x16 activation tile into LDS (shared by all 49 M-tiles).
  const float* tb = t + (size_t)b * RED * HW + p0;
  for (int i = tid; i < RED * 16; i += 256)
    t_lds[i] = tb[(size_t)(i >> 4) * HW + (i & 15)];
  __syncthreads();

  // GEMM: 49 M-tiles round-robined over 8 waves, K-loop 64/4 = 16 WMMAs.
  const int khalf = (lane >> 4) << 1;
  for (int mt = wave; mt < KK; mt += 8) {     // wave-uniform -> EXEC stays full
    const int row = mt * 16 + l15;
    const float* wrow = w2 + row * RED;
    v8f acc = {};
    #pragma unroll
    for (int k = 0; k < RED; k += 4) {
      const float2 av = *(const float2*)(wrow + k + khalf);
      v2f A; A.x = av.x; A.y = av.y;
      v2f B;
      B.x = t_lds[(k + khalf)     * 16 + l15];
      B.y = t_lds[(k + khalf + 1) * 16 + l15];
      acc = __builtin_amdgcn_wmma_f32_16x16x4_f32(
          false, A, false, B, (short)0, acc, false, false);
    }
    #pragma unroll
    for (int r = 0; r < 8; ++r) {
      const int o = mt * 16 + r + ((lane >> 4) << 3);
      wgt_lds[o * 16 + l15] = acc[r] + b2[o];
    }
  }
  __syncthreads();

  // Involution: 256 threads = 16 pixels (fast dim, coalesced x) x 16 groups.
  const int p  = tid & 15;
  const int g  = tid >> 4;
  const int hw = p0 + p;
  const int yy0 = hw / W56;
  const int xx0 = hw - yy0 * W56;

  float wv[KK];                               // this thread's 49 group weights
  #pragma unroll
  for (int k = 0; k < KK; ++k) wv[k] = wgt_lds[(g * KK + k) * 16 + p];

  const float* xg = x + ((size_t)b * CIN + g * GC) * HW;
  for (int j = 0; j < GC; ++j) {              // reuse wv over the 16 channels
    const int co = g * GC + j;
    const float* xc = xg + (size_t)j * HW;
    float acc = 0.f;
    #pragma unroll
    for (int di = 0; di < 7; ++di) {
      const int yy = yy0 + di - 3;
      if (yy < 0 || yy >= W56) continue;      // zero padding
      const float* xr = xc + yy * W56;
      #pragma unroll
      for (int dj = 0; dj < 7; ++dj) {
        const int xx = xx0 + dj - 3;
        if (xx >= 0 && xx < W56) acc = fmaf(wv[di * 7 + dj], xr[xx], acc);
      }
    }
    const float inv = g2[co] * rsqrtf(va2[co] + BN_EPS);
    const float val = fmaxf(acc * inv + (be2[co] - mu2[co] * inv), 0.f);
    out[((size_t)b * CIN + co) * HW + hw] = val;
  }
}

// ---------------------------------------------------------------------------
extern "C" void kernel_launch(void* const* d_in, const int* in_sizes, int n_in,
                              void* d_out, int out_size, void* d_ws, size_t ws_size,
                              hipStream_t stream) {
  const float* x    = (const float*)d_in[0];
  const float* w1   = (const float*)d_in[1];
  const float* g1   = (const float*)d_in[2];
  const float* be1  = (const float*)d_in[3];
  const float* mu1  = (const float*)d_in[4];
  const float* va1  = (const float*)d_in[5];
  const float* w2   = (const float*)d_in[6];
  const float* b2   = (const float*)d_in[7];
  const float* g2   = (const float*)d_in[8];
  const float* be2  = (const float*)d_in[9];
  const float* mu2  = (const float*)d_in[10];
  const float* va2  = (const float*)d_in[11];
  float* out = (float*)d_out;
  float* t   = (float*)d_ws;                  // [8, 64, 3136] fp32 = 6.4 MB

  dim3 grid1(HW / 32, 8);                     // 98 x 8 blocks, 256 thr (8 waves)
  conv1_bn_relu_kernel<<<grid1, 256, 0, stream>>>(x, w1, g1, be1, mu1, va1, t);

  dim3 grid2(HW / 16, 8);                     // 196 x 8 blocks, 256 thr
  conv2_involution_kernel<<<grid2, 256, 0, stream>>>(x, t, w2, b2, g2, be2, mu2,
                                                     va2, out);
  (void)in_sizes; (void)n_in; (void)out_size; (void)ws_size;
}